// TSModel_15848429322616
// MI455X (gfx1250) — compile-verified
//
#include <hip/hip_runtime.h>

typedef __attribute__((ext_vector_type(2))) float v2f;
typedef __attribute__((ext_vector_type(8))) float v8f;

constexpr int TT = 64;   // timesteps == layers
constexpr int BB = 128;  // batch
constexpr int HH = 64;   // hidden
constexpr int NSPLIT = 2;       // workgroups per cell (split within-gate columns)
constexpr int XSTRIDE = 132;    // LDS row stride (mod 64 == 4 -> conflict-free b64 loads)

__device__ __forceinline__ float fsig(float x)  { return 1.0f / (1.0f + __expf(-x)); }
__device__ __forceinline__ float ftanh(float x) { return 2.0f / (1.0f + __expf(-2.0f * x)) - 1.0f; }

// One wavefront d: all cells (t, l) with t + l == d. 2 WGs per cell, 256 thr each.
// GEMM: Z = [h_in | h_prev] (128x128) @ W (128x256), fused LSTM epilogue.
// X and the WG's weight tile (transposed) are both staged in LDS so the inner
// loop is pure ds_load_b64 + v_wmma with no VMEM latency on the wavefront path.
__launch_bounds__(256, 2)
__global__ void lstm_wavefront(
    const float* __restrict__ x,
    const float* __restrict__ W0, const float* __restrict__ b0,
    const float* __restrict__ Wl, const float* __restrict__ bl,
    float* __restrict__ hbuf, float* __restrict__ cbuf,   // [2][TT][BB][HH] each
    float* __restrict__ outs,                             // [TT][BB][HH]
    int dwave, int l_lo)
{
    __shared__ float Xs[BB * XSTRIDE];    // X = [h_in | h_prev], row-major
    __shared__ float Wt[128 * XSTRIDE];   // weight tile, TRANSPOSED: Wt[col_local][k_row]

    const int cell = blockIdx.x >> 1;
    const int s    = blockIdx.x & 1;          // within-gate column split (0 or 1)
    const int l    = l_lo + cell;
    const int t    = dwave - l;
    const int pr   = (dwave - 1) & 1;         // read parity
    const int pw   = dwave & 1;               // write parity
    const int tid  = threadIdx.x;

    const float* hin  = (l > 0) ? (hbuf + ((size_t)pr * TT + (l - 1)) * BB * HH) : nullptr;
    const float* hprv = hbuf + ((size_t)pr * TT + l) * BB * HH;
    const float* cprv = cbuf + ((size_t)pr * TT + l) * BB * HH;
    float*       hout = hbuf + ((size_t)pw * TT + l) * BB * HH;
    float*       cout = cbuf + ((size_t)pw * TT + l) * BB * HH;

    const float* W    = (l == 0) ? (W0 + 256) : (Wl + (size_t)(l - 1) * 128 * 256);
    const float* bias = (l == 0) ? b0 : (bl + (size_t)(l - 1) * 256);
    const int    kmax  = (l == 0) ? 16 : 32;  // layer 0: K=64 (input handled as rank-1)
    const int    Krows = kmax * 4;

    // ---- stage X = [h_in | h_prev] into LDS (zeros where state not yet defined)
    for (int i = 0; i < 16; ++i) {
        int c4  = tid + i * 256;              // float4 chunk id, 0..4095
        int row = c4 >> 5;                    // 32 float4 per row
        int col = (c4 & 31) * 4;
        float4 v = make_float4(0.f, 0.f, 0.f, 0.f);
        if (l == 0) {
            if (col < HH && t > 0) v = *(const float4*)(hprv + row * HH + col);
        } else {
            if (col < HH)          v = *(const float4*)(hin  + row * HH + col);
            else if (t > 0)        v = *(const float4*)(hprv + row * HH + (col - HH));
        }
        *(float4*)(&Xs[row * XSTRIDE + col]) = v;
    }

    // ---- stage weight tile transposed: Wt[g*32 + c][row] = W[row][g*64 + s*32 + c]
    // float4-coalesced global reads; scattered (padded-stride) LDS writes.
    {
        const int nchunks = Krows * 32;       // float4 chunks (128 cols / 4 per row)
        for (int e4 = tid; e4 < nchunks; e4 += 256) {
            int row = e4 >> 5;                // k row
            int cl4 = (e4 & 31) << 2;         // local col (multiple of 4)
            int g   = cl4 >> 5;               // gate 0..3
            int c   = cl4 & 31;               // col within gate-half
            float4 wv = *(const float4*)(W + (size_t)row * 256 + g * 64 + s * 32 + c);
            Wt[(cl4 + 0) * XSTRIDE + row] = wv.x;
            Wt[(cl4 + 1) * XSTRIDE + row] = wv.y;
            Wt[(cl4 + 2) * XSTRIDE + row] = wv.z;
            Wt[(cl4 + 3) * XSTRIDE + row] = wv.w;
        }
    }
    __syncthreads();

    const int wave = tid >> 5;
    const int lane = tid & 31;
    const int m0   = wave * 16;               // 8 waves x 16 batch rows
    const int lm   = lane & 15;
    const int kk   = (lane >> 4) << 1;        // K sub-offset per A/B fragment layout

    for (int qq = 0; qq < 2; ++qq) {
        const int q    = s * 2 + qq;          // within-gate 16-col tile, 0..3
        const int ncol = q * 16 + lm;         // within-gate column 0..63 (global)
        const int clb  = qq * 16 + lm;        // local column base inside Wt gate block

        const float* wti = &Wt[(0 * 32 + clb) * XSTRIDE];
        const float* wtj = &Wt[(1 * 32 + clb) * XSTRIDE];
        const float* wtf = &Wt[(2 * 32 + clb) * XSTRIDE];
        const float* wto = &Wt[(3 * 32 + clb) * XSTRIDE];
        const float* xr0 = &Xs[(m0 + lm) * XSTRIDE];

        v8f acc_i = {}, acc_j = {}, acc_f = {}, acc_o = {};
        for (int k = 0; k < kmax; ++k) {
            const int k4 = k * 4 + kk;
            v2f a;  a.x  = xr0[k4];       a.y  = xr0[k4 + 1];
            v2f bi; bi.x = wti[k4];       bi.y = wti[k4 + 1];
            v2f bj; bj.x = wtj[k4];       bj.y = wtj[k4 + 1];
            v2f bf; bf.x = wtf[k4];       bf.y = wtf[k4 + 1];
            v2f bo; bo.x = wto[k4];       bo.y = wto[k4 + 1];
            acc_i = __builtin_amdgcn_wmma_f32_16x16x4_f32(false, a, false, bi, (short)0, acc_i, false, false);
            acc_j = __builtin_amdgcn_wmma_f32_16x16x4_f32(false, a, false, bj, (short)0, acc_j, false, false);
            acc_f = __builtin_amdgcn_wmma_f32_16x16x4_f32(false, a, false, bf, (short)0, acc_f, false, false);
            acc_o = __builtin_amdgcn_wmma_f32_16x16x4_f32(false, a, false, bo, (short)0, acc_o, false, false);
        }

        // ---- fused LSTM epilogue
        const float bi_b = bias[ncol];
        const float bj_b = bias[64 + ncol];
        const float bf_b = bias[128 + ncol];
        const float bo_b = bias[192 + ncol];
        float w0i = 0.f, w0j = 0.f, w0f = 0.f, w0o = 0.f;
        if (l == 0) {
            w0i = W0[ncol]; w0j = W0[64 + ncol]; w0f = W0[128 + ncol]; w0o = W0[192 + ncol];
        }
        #pragma unroll
        for (int r = 0; r < 8; ++r) {
            const int b = m0 + r + ((lane >> 4) << 3);   // batch row of this C/D element
            float zi = acc_i[r] + bi_b;
            float zj = acc_j[r] + bj_b;
            float zf = acc_f[r] + bf_b;
            float zo = acc_o[r] + bo_b;
            if (l == 0) {
                const float xv = x[b * TT + t];
                zi += xv * w0i; zj += xv * w0j; zf += xv * w0f; zo += xv * w0o;
            }
            const float cp = (t > 0) ? cprv[b * HH + ncol] : 0.0f;
            const float c2 = cp * fsig(zf) + fsig(zi) * ftanh(zj);
            const float h2 = ftanh(c2) * fsig(zo);
            cout[b * HH + ncol] = c2;
            hout[b * HH + ncol] = h2;
            if (l == TT - 1) outs[((size_t)t * BB + b) * HH + ncol] = h2;
        }
    }
}

// pred[b][t] = relu(outs[t][b][:] . Wd[t][:,0] + bd[t])
__global__ void dense_relu(const float* __restrict__ outs, const float* __restrict__ Wd,
                           const float* __restrict__ bd, float* __restrict__ pred)
{
    int idx = blockIdx.x * 256 + threadIdx.x;   // idx = b*TT + t
    if (idx >= BB * TT) return;
    int b = idx >> 6;
    int t = idx & 63;
    const float* op = outs + ((size_t)t * BB + b) * HH;
    const float* wp = Wd + t * HH;
    float acc = bd[t];
    #pragma unroll
    for (int h = 0; h < HH; ++h) acc += op[h] * wp[h];
    pred[idx] = fmaxf(acc, 0.0f);
}

// loss = mean((labels - pred)^2), single WG deterministic tree reduction
__global__ void mse_loss(const float* __restrict__ labels, const float* __restrict__ pred,
                         float* __restrict__ loss)
{
    __shared__ float red[256];
    int tid = threadIdx.x;
    float s = 0.f;
    for (int i = tid; i < BB * TT; i += 256) {
        float d = labels[i] - pred[i];
        s += d * d;
    }
    red[tid] = s;
    __syncthreads();
    for (int w = 128; w > 0; w >>= 1) {
        if (tid < w) red[tid] += red[tid + w];
        __syncthreads();
    }
    if (tid == 0) loss[0] = red[0] / (float)(BB * TT);
}

extern "C" void kernel_launch(void* const* d_in, const int* in_sizes, int n_in,
                              void* d_out, int out_size, void* d_ws, size_t ws_size,
                              hipStream_t stream)
{
    const float* x      = (const float*)d_in[0];  // [B,T,1]
    const float* labels = (const float*)d_in[1];  // [B,T,1]
    const float* W0     = (const float*)d_in[2];  // [65,256]
    const float* b0     = (const float*)d_in[3];  // [256]
    const float* Wl     = (const float*)d_in[4];  // [63,128,256]
    const float* bl     = (const float*)d_in[5];  // [63,256]
    const float* Wd     = (const float*)d_in[6];  // [64,64,1]
    const float* bd     = (const float*)d_in[7];  // [64,1]

    float* pred = (float*)d_out;                  // 8192
    float* loss = pred + BB * TT;                 // +1

    float* ws   = (float*)d_ws;
    float* hbuf = ws;                             // 2*64*128*64
    float* cbuf = hbuf + 2 * TT * BB * HH;        // 2*64*128*64
    float* outs = cbuf + 2 * TT * BB * HH;        // 64*128*64

    // 127 anti-diagonal wavefronts; kernel boundaries provide the (t,l) deps.
    for (int d = 0; d < 2 * TT - 1; ++d) {
        int l_lo = (d > TT - 1) ? (d - (TT - 1)) : 0;
        int l_hi = (d < TT - 1) ? d : (TT - 1);
        int ncells = l_hi - l_lo + 1;
        lstm_wavefront<<<dim3(ncells * NSPLIT), dim3(256), 0, stream>>>(
            x, W0, b0, Wl, bl, hbuf, cbuf, outs, d, l_lo);
    }
    dense_relu<<<dim3((BB * TT + 255) / 256), dim3(256), 0, stream>>>(outs, Wd, bd, pred);
    mse_loss<<<dim3(1), dim3(256), 0, stream>>>(labels, pred, loss);
}